// RGCNGRU_78254304133682
// MI455X (gfx1250) — compile-verified
//
#include <hip/hip_runtime.h>
#include <math.h>

// ---------------- problem constants (match reference) ----------------
#define N_NODES 100000
#define N_EDGES 1000000
#define D 64
#define NUM_REL 3
#define BN_EPS 1e-5f

typedef __attribute__((ext_vector_type(2))) float v2f;
typedef __attribute__((ext_vector_type(8))) float v8f;

// =====================================================================
// Kernel: zero a float buffer
// =====================================================================
__global__ void zero_kernel(float* __restrict__ p, int n) {
    int gid = blockIdx.x * blockDim.x + threadIdx.x;
    if (gid < n) p[gid] = 0.0f;
}

// =====================================================================
// Kernel: t[r, n, e] = sum_k H[n, k] * W[r, k, e]
// One wave computes a 16-row x 64-col strip for one relation r, using
// V_WMMA_F32_16X16X4_F32 (fp32 WMMA, K=4), 64 WMMAs per wave.
//
// A (16x4 f32) per-lane: row = lane%16, VGPR v holds K = (lane<16?0:2)+v
// B (4x16 f32) per-lane: col = lane%16, VGPR v holds K = (lane<16?0:2)+v
// C/D (16x16 f32): VGPR v, lanes 0-15 -> M=v, lanes 16-31 -> M=v+8, N=lane%16
// =====================================================================
__global__ void transform_kernel(const float* __restrict__ H,
                                 const float* __restrict__ W,
                                 float* __restrict__ T) {
    const int wavesPerBlock = blockDim.x >> 5;
    const int job  = blockIdx.x * wavesPerBlock + (threadIdx.x >> 5);
    const int lane = threadIdx.x & 31;
    const int nJobs = (N_NODES / 16) * NUM_REL;
    if (job >= nJobs) return;               // wave-uniform: EXEC stays all-1s

    const int r       = job % NUM_REL;
    const int rowTile = job / NUM_REL;
    const int rowbase = rowTile * 16;
    const int lmod    = lane & 15;
    const int khalf   = (lane < 16) ? 0 : 2;

    const float* __restrict__ hrow = H + (size_t)(rowbase + lmod) * D;
    const float* __restrict__ Wr   = W + (size_t)r * D * D;

    v8f acc0 = {}, acc1 = {}, acc2 = {}, acc3 = {};
    #pragma unroll
    for (int k0 = 0; k0 < D; k0 += 4) {
        const int k = k0 + khalf;
        v2f a; a.x = hrow[k]; a.y = hrow[k + 1];
        v2f b0; b0.x = Wr[(size_t)k * D +  0 + lmod]; b0.y = Wr[(size_t)(k + 1) * D +  0 + lmod];
        v2f b1; b1.x = Wr[(size_t)k * D + 16 + lmod]; b1.y = Wr[(size_t)(k + 1) * D + 16 + lmod];
        v2f b2; b2.x = Wr[(size_t)k * D + 32 + lmod]; b2.y = Wr[(size_t)(k + 1) * D + 32 + lmod];
        v2f b3; b3.x = Wr[(size_t)k * D + 48 + lmod]; b3.y = Wr[(size_t)(k + 1) * D + 48 + lmod];
        acc0 = __builtin_amdgcn_wmma_f32_16x16x4_f32(false, a, false, b0, (short)0, acc0, false, false);
        acc1 = __builtin_amdgcn_wmma_f32_16x16x4_f32(false, a, false, b1, (short)0, acc1, false, false);
        acc2 = __builtin_amdgcn_wmma_f32_16x16x4_f32(false, a, false, b2, (short)0, acc2, false, false);
        acc3 = __builtin_amdgcn_wmma_f32_16x16x4_f32(false, a, false, b3, (short)0, acc3, false, false);
    }

    const int mbase = rowbase + ((lane < 16) ? 0 : 8);
    float* __restrict__ Tr = T + (size_t)r * N_NODES * D;
    #pragma unroll
    for (int v = 0; v < 8; ++v) {
        float* row = Tr + (size_t)(mbase + v) * D + lmod;
        row[ 0] = acc0[v];
        row[16] = acc1[v];
        row[32] = acc2[v];
        row[48] = acc3[v];
    }
}

// =====================================================================
// Kernel: per-edge gather from t + norm scale + atomic scatter-add to swh
// 16 lanes per edge, 4 floats per lane (float4 load, 4 f32 atomics).
// Entire t (76.8MB) and swh (25.6MB) are L2-resident (192MB L2).
// =====================================================================
__global__ void edge_scatter_kernel(const float* __restrict__ T,
                                    const float* __restrict__ norm,
                                    const int* __restrict__ src,
                                    const int* __restrict__ dst,
                                    const int* __restrict__ rel,
                                    float* __restrict__ swh) {
    const int gid  = blockIdx.x * blockDim.x + threadIdx.x;
    const int e    = gid >> 4;
    const int part = gid & 15;
    if (e >= N_EDGES) return;

    const int   r = rel[e];
    const int   s = src[e];
    const int   d = dst[e];
    const float w = norm[e];

    const float4 v = *(const float4*)(T + ((size_t)r * N_NODES + s) * D + part * 4);
    float* p = swh + (size_t)d * D + part * 4;
    atomicAdd(p + 0, v.x * w);
    atomicAdd(p + 1, v.y * w);
    atomicAdd(p + 2, v.z * w);
    atomicAdd(p + 3, v.w * w);
}

// =====================================================================
// Kernel: GRU cell. One wave computes a 16-node x 16-feature output tile.
// gi = x @ w_ih.T   -> B[k][g] = w_ih[g*64 + k]   (g = j, j+64, j+128)
// gh = h @ w_hh.T   -> B[k][g] = w_hh[g*64 + k]
// 6 accumulators (r/z/n for input and hidden), 96 WMMAs, fused gates.
// Double-buffered h (hin -> hout), so feature-split waves are race-free.
// =====================================================================
__device__ __forceinline__ float sigmoidf_(float x) {
    return 1.0f / (1.0f + __expf(-x));
}

__global__ void gru_kernel(const float* __restrict__ swh,
                           const float* __restrict__ hin,
                           float* __restrict__ hout,
                           const float* __restrict__ w_ih,
                           const float* __restrict__ w_hh,
                           const float* __restrict__ b_ih,
                           const float* __restrict__ b_hh) {
    const int wavesPerBlock = blockDim.x >> 5;
    const int job  = blockIdx.x * wavesPerBlock + (threadIdx.x >> 5);
    const int lane = threadIdx.x & 31;
    const int nJobs = (N_NODES / 16) * (D / 16);
    if (job >= nJobs) return;               // wave-uniform

    const int nodeTile = job >> 2;          // D/16 == 4 feature tiles
    const int jTile    = job & 3;
    const int rowbase  = nodeTile * 16;
    const int jbase    = jTile * 16;
    const int lmod     = lane & 15;
    const int khalf    = (lane < 16) ? 0 : 2;
    const int j        = jbase + lmod;

    const float* __restrict__ xrow = swh + (size_t)(rowbase + lmod) * D;
    const float* __restrict__ hrow = hin + (size_t)(rowbase + lmod) * D;

    v8f ir = {}, iz = {}, in_ = {}, hr = {}, hz = {}, hn = {};
    #pragma unroll
    for (int k0 = 0; k0 < D; k0 += 4) {
        const int k = k0 + khalf;
        v2f ax; ax.x = xrow[k]; ax.y = xrow[k + 1];
        v2f ah; ah.x = hrow[k]; ah.y = hrow[k + 1];

        v2f bri; bri.x = w_ih[(size_t)(j      ) * D + k]; bri.y = w_ih[(size_t)(j      ) * D + k + 1];
        v2f bzi; bzi.x = w_ih[(size_t)(j +  64) * D + k]; bzi.y = w_ih[(size_t)(j +  64) * D + k + 1];
        v2f bni; bni.x = w_ih[(size_t)(j + 128) * D + k]; bni.y = w_ih[(size_t)(j + 128) * D + k + 1];
        v2f brh; brh.x = w_hh[(size_t)(j      ) * D + k]; brh.y = w_hh[(size_t)(j      ) * D + k + 1];
        v2f bzh; bzh.x = w_hh[(size_t)(j +  64) * D + k]; bzh.y = w_hh[(size_t)(j +  64) * D + k + 1];
        v2f bnh; bnh.x = w_hh[(size_t)(j + 128) * D + k]; bnh.y = w_hh[(size_t)(j + 128) * D + k + 1];

        ir  = __builtin_amdgcn_wmma_f32_16x16x4_f32(false, ax, false, bri, (short)0, ir,  false, false);
        iz  = __builtin_amdgcn_wmma_f32_16x16x4_f32(false, ax, false, bzi, (short)0, iz,  false, false);
        in_ = __builtin_amdgcn_wmma_f32_16x16x4_f32(false, ax, false, bni, (short)0, in_, false, false);
        hr  = __builtin_amdgcn_wmma_f32_16x16x4_f32(false, ah, false, brh, (short)0, hr,  false, false);
        hz  = __builtin_amdgcn_wmma_f32_16x16x4_f32(false, ah, false, bzh, (short)0, hz,  false, false);
        hn  = __builtin_amdgcn_wmma_f32_16x16x4_f32(false, ah, false, bnh, (short)0, hn,  false, false);
    }

    const float bir = b_ih[j], biz = b_ih[j + 64], bin = b_ih[j + 128];
    const float bhr = b_hh[j], bhz = b_hh[j + 64], bhn = b_hh[j + 128];
    const int mbase = rowbase + ((lane < 16) ? 0 : 8);

    #pragma unroll
    for (int v = 0; v < 8; ++v) {
        const int m = mbase + v;
        const float hold = hin[(size_t)m * D + j];
        const float rg = sigmoidf_((ir[v] + bir) + (hr[v] + bhr));
        const float zg = sigmoidf_((iz[v] + biz) + (hz[v] + bhz));
        const float ng = tanhf((in_[v] + bin) + rg * (hn[v] + bhn));
        hout[(size_t)m * D + j] = (1.0f - zg) * ng + zg * hold;
    }
}

// =====================================================================
// BatchNorm: stats (atomic sum / sumsq per feature), then apply.
// =====================================================================
__global__ void bn_stats_kernel(const float* __restrict__ H,
                                float* __restrict__ stats) {
    const int t  = threadIdx.x;
    const int jf = t & 63;
    const int rq = t >> 6;                  // 0..3 with 256 threads
    float s = 0.0f, sq = 0.0f;
    for (int row = blockIdx.x * 4 + rq; row < N_NODES; row += gridDim.x * 4) {
        const float v = H[(size_t)row * D + jf];
        s  += v;
        sq += v * v;
    }
    atomicAdd(&stats[jf], s);
    atomicAdd(&stats[64 + jf], sq);
}

__global__ void bn_apply_kernel(const float* __restrict__ H,
                                const float* __restrict__ stats,
                                const float* __restrict__ gamma,
                                const float* __restrict__ beta,
                                float* __restrict__ out) {
    const int gid = blockIdx.x * blockDim.x + threadIdx.x;
    if (gid >= N_NODES * D) return;
    const int jf = gid & 63;
    const float invN = 1.0f / (float)N_NODES;
    const float mean = stats[jf] * invN;
    const float var  = stats[64 + jf] * invN - mean * mean;
    out[gid] = (H[gid] - mean) * rsqrtf(var + BN_EPS) * gamma[jf] + beta[jf];
}

// =====================================================================
// Host orchestration
// =====================================================================
extern "C" void kernel_launch(void* const* d_in, const int* in_sizes, int n_in,
                              void* d_out, int out_size, void* d_ws, size_t ws_size,
                              hipStream_t stream) {
    const float* h      = (const float*)d_in[0];
    const float* norm   = (const float*)d_in[1];
    const float* weight = (const float*)d_in[2];
    const float* w_ih   = (const float*)d_in[3];
    const float* w_hh   = (const float*)d_in[4];
    const float* b_ih   = (const float*)d_in[5];
    const float* b_hh   = (const float*)d_in[6];
    const float* gamma  = (const float*)d_in[7];
    const float* beta   = (const float*)d_in[8];
    const int*   src    = (const int*)d_in[9];
    const int*   dst    = (const int*)d_in[10];
    const int*   rel    = (const int*)d_in[11];

    // workspace layout (floats)
    char* ws = (char*)d_ws;
    const size_t T_BYTES  = (size_t)NUM_REL * N_NODES * D * sizeof(float); // 76.8 MB
    const size_t ND_BYTES = (size_t)N_NODES * D * sizeof(float);           // 25.6 MB
    float* T     = (float*)(ws);
    float* swh   = (float*)(ws + T_BYTES);
    float* h0    = (float*)(ws + T_BYTES + ND_BYTES);
    float* h1    = (float*)(ws + T_BYTES + 2 * ND_BYTES);
    float* stats = (float*)(ws + T_BYTES + 3 * ND_BYTES);

    const int ND = N_NODES * D;

    // init: hstate = 0, bn stats = 0
    zero_kernel<<<(ND + 255) / 256, 256, 0, stream>>>(h0, ND);
    zero_kernel<<<1, 128, 0, stream>>>(stats, 128);

    // transform grid: (N/16 row tiles) * NUM_REL waves, 8 waves per block
    const int tJobs   = (N_NODES / 16) * NUM_REL;
    const int tBlocks = (tJobs + 7) / 8;
    // edge scatter grid: 16 lanes per edge
    const int eBlocks = (N_EDGES * 16 + 255) / 256;
    // gru grid: (N/16) * (D/16) waves, 8 waves per block
    const int gJobs   = (N_NODES / 16) * (D / 16);
    const int gBlocks = (gJobs + 7) / 8;

    // ---------------- layer 1 ----------------
    transform_kernel<<<tBlocks, 256, 0, stream>>>(h, weight, T);
    zero_kernel<<<(ND + 255) / 256, 256, 0, stream>>>(swh, ND);
    edge_scatter_kernel<<<eBlocks, 256, 0, stream>>>(T, norm, src, dst, rel, swh);
    gru_kernel<<<gBlocks, 256, 0, stream>>>(swh, h0, h1, w_ih, w_hh, b_ih, b_hh);

    // ---------------- layer 2 ----------------
    transform_kernel<<<tBlocks, 256, 0, stream>>>(h1, weight, T);
    zero_kernel<<<(ND + 255) / 256, 256, 0, stream>>>(swh, ND);
    edge_scatter_kernel<<<eBlocks, 256, 0, stream>>>(T, norm, src, dst, rel, swh);
    gru_kernel<<<gBlocks, 256, 0, stream>>>(swh, h1, h0, w_ih, w_hh, b_ih, b_hh);

    // ---------------- batchnorm ----------------
    bn_stats_kernel<<<512, 256, 0, stream>>>(h0, stats);
    bn_apply_kernel<<<(ND + 255) / 256, 256, 0, stream>>>(h0, stats, gamma, beta, (float*)d_out);
}